// ppro_cd_loss_88038239634155
// MI455X (gfx1250) — compile-verified
//
#include <hip/hip_runtime.h>

typedef __attribute__((ext_vector_type(2))) float v2f;
typedef __attribute__((ext_vector_type(8))) float v8f;

#define BATCH 8
#define NPTS 8192
#define HALF 4096
#define TILES_PER_HALF 256     // 4096 points / 16 cols per tile
#define ROWS_PER_BLOCK 128     // 8 waves * 16 rows

// For each (b, n): out[b*NPTS + n] = min_m || pa[b,n] - pb[b,m] ||^2
__global__ __launch_bounds__(256)
void chamfer_rowmin_kernel(const float* __restrict__ pa,
                           const float* __restrict__ pb,
                           float* __restrict__ out)
{
    // B operand staged in WMMA lane layout: entry = tile*32 + lane
    __shared__ v2f ldsB[TILES_PER_HALF * 32];   // 64 KB

    const int b    = blockIdx.y;
    const int lane = threadIdx.x & 31;          // wave32
    const int wave = threadIdx.x >> 5;
    const int rowBase = blockIdx.x * ROWS_PER_BLOCK + wave * 16;
    const int r = lane & 15;                    // my A row within the 16-row tile
    const int n = rowBase + r;

    // ---- A operand: 16x4 f32, row n = (ax, ay, az, 1)
    // lanes 0-15 hold (K0,K1)=(x,y); lanes 16-31 hold (K2,K3)=(z,1)
    const float* ap = pa + ((size_t)b * NPTS + n) * 3;
    const float ax = ap[0], ay = ap[1], az = ap[2];
    const float a2 = ax * ax + ay * ay + az * az;
    v2f aop;
    if (lane < 16) { aop.x = ax; aop.y = ay; }
    else           { aop.x = az; aop.y = 1.0f; }

    v8f rowmin;
#pragma unroll
    for (int i = 0; i < 8; ++i) rowmin[i] = 3.0e38f;
    v8f czero = {};

    for (int h = 0; h < 2; ++h) {
        // ---- stage B' for 4096 points: col m = (-2bx,-2by,-2bz, b2)
        for (int e = threadIdx.x; e < TILES_PER_HALF * 32; e += 256) {
            const int tile = e >> 5;
            const int el   = e & 31;
            const int m    = h * HALF + tile * 16 + (el & 15);
            const float* bp = pb + ((size_t)b * NPTS + m) * 3;
            const float bx = bp[0], by = bp[1], bz = bp[2];
            v2f v;
            if (el < 16) { v.x = -2.0f * bx; v.y = -2.0f * by; }
            else         { v.x = -2.0f * bz; v.y = bx * bx + by * by + bz * bz; }
            ldsB[e] = v;
        }
        __syncthreads();

        // ---- sweep 256 column tiles: 1 WMMA + 8 v_min per 256 pairs
#pragma unroll 4
        for (int t = 0; t < TILES_PER_HALF; ++t) {
            v2f bop = ldsB[t * 32 + lane];
            // D = A x B + 0 ;  D[n][m] = -2 a.b + ||b||^2
            v8f d = __builtin_amdgcn_wmma_f32_16x16x4_f32(
                false, aop, false, bop, (short)0, czero, false, false);
#pragma unroll
            for (int i = 0; i < 8; ++i)
                rowmin[i] = __builtin_fminf(rowmin[i], d[i]);
        }
        __syncthreads();
    }

    // ---- add ||a||^2 per row: D-layout row of VGPR i = i + 8*(lane>=16).
    // a2 lives at lane (row & 15); gather per VGPR via shuffle.
    const int selHi = (lane >> 4) << 3;
#pragma unroll
    for (int i = 0; i < 8; ++i) {
        const float a2row = __shfl(a2, i + selHi, 32);
        rowmin[i] += a2row;
    }

    // ---- min across the 16 columns (16-lane half-wave xor reduction)
#pragma unroll
    for (int off = 1; off <= 8; off <<= 1) {
#pragma unroll
        for (int i = 0; i < 8; ++i)
            rowmin[i] = __builtin_fminf(rowmin[i], __shfl_xor(rowmin[i], off, 32));
    }

    // lanes 0 / 16 hold mins for rows i / i+8
    if (lane == 0) {
#pragma unroll
        for (int i = 0; i < 8; ++i)
            out[(size_t)b * NPTS + rowBase + i] = rowmin[i];
    } else if (lane == 16) {
#pragma unroll
        for (int i = 0; i < 8; ++i)
            out[(size_t)b * NPTS + rowBase + 8 + i] = rowmin[i];
    }
}

// Deterministic single-block reduction: out = sum(ws[0 : 2*B*N]) / (B*N)
// (mean(d1) + mean(d2) == total_sum / 65536 since N == M)
__global__ __launch_bounds__(256)
void chamfer_reduce_kernel(const float* __restrict__ ws, float* __restrict__ out)
{
    __shared__ double sdata[256];
    double s = 0.0;
    const int total = 2 * BATCH * NPTS;
    for (int i = threadIdx.x; i < total; i += 256) s += (double)ws[i];
    sdata[threadIdx.x] = s;
    __syncthreads();
    for (int k = 128; k > 0; k >>= 1) {
        if (threadIdx.x < k) sdata[threadIdx.x] += sdata[threadIdx.x + k];
        __syncthreads();
    }
    if (threadIdx.x == 0)
        out[0] = (float)(sdata[0] / (double)(BATCH * NPTS));
}

extern "C" void kernel_launch(void* const* d_in, const int* in_sizes, int n_in,
                              void* d_out, int out_size, void* d_ws, size_t ws_size,
                              hipStream_t stream)
{
    (void)in_sizes; (void)n_in; (void)out_size; (void)ws_size;
    const float* p1 = (const float*)d_in[0];
    const float* p2 = (const float*)d_in[1];
    float* ws  = (float*)d_ws;                 // 2 * 8 * 8192 floats = 512 KB
    float* out = (float*)d_out;

    dim3 grid(NPTS / ROWS_PER_BLOCK, BATCH);   // (64, 8)
    // direction 1: for each p1 point, nearest in p2
    chamfer_rowmin_kernel<<<grid, 256, 0, stream>>>(p1, p2, ws);
    // direction 2: swapped
    chamfer_rowmin_kernel<<<grid, 256, 0, stream>>>(p2, p1, ws + (size_t)BATCH * NPTS);
    // final scalar
    chamfer_reduce_kernel<<<1, 256, 0, stream>>>(ws, out);
}